// Spatial_69913477644421
// MI455X (gfx1250) — compile-verified
//
#include <hip/hip_runtime.h>
#include <math.h>

// fused: LayerNorm(A=128) -> W(256x256) @ G(256x128) + bias -> GELU(erf, branch-free) -> +residual
// one workgroup (256 threads = 8 wave32) per batch element; G staged in LDS in K-pair-interleaved
// layout so WMMA B fragments are single ds_load_b64; GEMM on V_WMMA_F32_16X16X4_F32.

typedef __attribute__((ext_vector_type(2))) float v2f;
typedef __attribute__((ext_vector_type(8))) float v8f;

#define KK_DIM 256
#define A_DIM 128
#define ROW_ELEMS (KK_DIM * A_DIM)     // 32768 floats per batch element
// LDS layout: pair-row p (= k/2) holds 128 cols x 2 k-parities interleaved:
//   addr = p * PAIR_STRIDE + col*2 + (k&1)
// PAIR_STRIDE % 64 == 32  ->  lane halves (which differ by one pair-row) hit
// disjoint bank sets for conflict-free ds_load_b64 B fetches.
#define PAIR_STRIDE 288
#define LN_EPS 1e-5f

// Branch-free exact-erf GELU: Abramowitz-Stegun 7.1.26 (|err| < 1.5e-7).
__device__ __forceinline__ float gelu_erf(float v) {
    const float z  = v * 0.70710678118654752f;   // x / sqrt(2)
    const float az = fabsf(z);
    const float t  = __builtin_amdgcn_rcpf(fmaf(0.3275911f, az, 1.0f));
    float p = fmaf(t, 1.061405429f, -1.453152027f);
    p = fmaf(t, p, 1.421413741f);
    p = fmaf(t, p, -0.284496736f);
    p = fmaf(t, p, 0.254829592f);
    p *= t;
    const float e = __expf(-az * az);            // v_exp_f32 path, no branches
    float erfv = fmaf(-p, e, 1.0f);
    erfv = copysignf(erfv, z);
    return 0.5f * v * (1.0f + erfv);
}

__launch_bounds__(256, 2)
__global__ void fused_ln_chanmix_gelu(const float* __restrict__ x,
                                      const float* __restrict__ w,
                                      const float* __restrict__ bconv,
                                      const float* __restrict__ gamma,
                                      const float* __restrict__ beta,
                                      float* __restrict__ out)
{
    __shared__ float g_lds[(KK_DIM / 2) * PAIR_STRIDE];   // 144 KB

    const int b    = blockIdx.x;
    const int tid  = threadIdx.x;
    const int wave = tid >> 5;       // 0..7
    const int lane = tid & 31;       // wave32 lane

    const float* xb = x + (size_t)b * ROW_ELEMS;

    // ---------------- Phase 1: LayerNorm over A, normalized G -> LDS ----------------
    // One 128-float row per wave iteration: 4 floats/lane (coalesced float4 load).
    const float4 gam4 = ((const float4*)gamma)[lane];
    const float4 bet4 = ((const float4*)beta)[lane];

    for (int j = wave; j < KK_DIM; j += 8) {
        float4 v = ((const float4*)(xb + j * A_DIM))[lane];
        float s  = v.x + v.y + v.z + v.w;
        float sq = v.x * v.x + v.y * v.y + v.z * v.z + v.w * v.w;
        // wave32 butterfly reduction (warpSize == 32 on gfx1250)
        #pragma unroll
        for (int off = 16; off > 0; off >>= 1) {
            s  += __shfl_xor(s,  off, 32);
            sq += __shfl_xor(sq, off, 32);
        }
        const float mean = s * (1.0f / 128.0f);
        const float var  = sq * (1.0f / 128.0f) - mean * mean;
        const float rstd = rsqrtf(var + LN_EPS);

        float4 g;
        g.x = (v.x - mean) * rstd * gam4.x + bet4.x;
        g.y = (v.y - mean) * rstd * gam4.y + bet4.y;
        g.z = (v.z - mean) * rstd * gam4.z + bet4.z;
        g.w = (v.w - mean) * rstd * gam4.w + bet4.w;

        // K-pair interleaved store: lds[(j>>1)*PAIR_STRIDE + a*2 + (j&1)]
        float* dst = &g_lds[(j >> 1) * PAIR_STRIDE + (lane * 4) * 2 + (j & 1)];
        dst[0] = g.x; dst[2] = g.y; dst[4] = g.z; dst[6] = g.w;
    }
    __syncthreads();

    // ---------------- Phase 2: Out = W @ G via V_WMMA_F32_16X16X4_F32 ----------------
    // A (16x4 f32) fragment: lanes 0-15 -> M=lane, lanes 16-31 -> M=lane-16;
    //   VGPR0 holds K = {0 | 2} (by lane half), VGPR1 holds K = {1 | 3}.
    // B (4x16 f32) fragment (symmetric): VGPR0/1 hold K = {0,1} for lanes 0-15
    //   and K = {2,3} for lanes 16-31 -> one ds_load_b64 per fragment in the
    //   pair-interleaved LDS layout.
    const int half = lane >> 4;           // 0: lanes 0-15, 1: lanes 16-31
    const int l15  = lane & 15;
    const int koff = half << 1;           // 0 or 2

    for (int mrep = 0; mrep < 2; ++mrep) {
        const int i0 = (wave * 2 + mrep) * 16;            // M-tile base row of W/out
        const float* wrow = w + (size_t)(i0 + l15) * KK_DIM;

        v8f acc[8];
        #pragma unroll
        for (int t = 0; t < 8; ++t)
            acc[t] = (v8f){0.f, 0.f, 0.f, 0.f, 0.f, 0.f, 0.f, 0.f};

        for (int jt = 0; jt < KK_DIM / 4; ++jt) {         // K loop, 4 per WMMA
            const int kb = jt * 4 + koff;                 // this lane-half's K base
            const v2f av = *(const v2f*)(wrow + kb);      // global_load_b64, A reused x8
            // pair-row index = kb>>1 = jt*2 + half
            const float* gb = &g_lds[(size_t)(jt * 2 + half) * PAIR_STRIDE];
            #pragma unroll
            for (int t = 0; t < 8; ++t) {                 // 8 N-tiles of 16 cols
                const v2f bv = *(const v2f*)(gb + (t * 16 + l15) * 2);  // ds_load_b64
                acc[t] = __builtin_amdgcn_wmma_f32_16x16x4_f32(
                    /*neg_a=*/false, av, /*neg_b=*/false, bv,
                    /*c_mod=*/(short)0, acc[t],
                    /*reuse_a=*/false, /*reuse_b=*/false);
            }
        }

        // Epilogue: bias + GELU + residual, store.
        // C/D layout: VGPR r -> M = r + 8*half, N = l15.
        const int mbase = half * 8;
        #pragma unroll
        for (int t = 0; t < 8; ++t) {
            const int a0 = t * 16;
            #pragma unroll
            for (int r = 0; r < 8; ++r) {
                const int i = i0 + mbase + r;
                float vv = acc[t][r] + bconv[i];
                float ge = gelu_erf(vv);
                const size_t off = (size_t)i * A_DIM + (a0 + l15);
                out[(size_t)b * ROW_ELEMS + off] = ge + xb[off];
            }
        }
    }
}

extern "C" void kernel_launch(void* const* d_in, const int* in_sizes, int n_in,
                              void* d_out, int out_size, void* d_ws, size_t ws_size,
                              hipStream_t stream) {
    (void)n_in; (void)out_size; (void)d_ws; (void)ws_size;
    const float* x     = (const float*)d_in[0];
    const float* w     = (const float*)d_in[1];
    const float* bconv = (const float*)d_in[2];
    const float* gamma = (const float*)d_in[3];
    const float* beta  = (const float*)d_in[4];
    float* out = (float*)d_out;

    const int batch = in_sizes[0] / ROW_ELEMS;   // 2048
    fused_ln_chanmix_gelu<<<batch, 256, 0, stream>>>(x, w, bconv, gamma, beta, out);
}